// OneLabelLPModel_85856396248058
// MI455X (gfx1250) — compile-verified
//
#include <hip/hip_runtime.h>
#include <cstdint>

// ---------------------------------------------------------------------------
// 2-layer sum-aggregation GNN (gather/scatter-add), memory-bound:
//   HBM floor ~512 MB (edge indices, read once per layer) -> ~22 us @ 23.3 TB/s.
// CDNA5 path: global_load_async_to_lds_b128 (ASYNCcnt-tracked, NT hint so the
// read-once index stream doesn't evict the L2-resident x/agg arrays), 4-stage
// per-wave circular pipeline (~24 MB aggregate in flight across 8192 waves,
// matching the HBM BW*latency product), s_wait_asynccnt fencing, ds_load_b128
// index reads, global_atomic_add_f32 scatters.
// ---------------------------------------------------------------------------

#define THREADS 256
#define WAVESZ  32
#define WPB     (THREADS / WAVESZ)    // 8 waves / block (wave32)
#define EPW     64                    // edges per wave per pipeline stage
#define BUF_BYTES 1024                // 512B src idx + 512B dst idx
#define PIPE    4                     // pipeline depth (power of two)
#define WAVE_LDS (PIPE * BUF_BYTES)   // 4 KB per wave

// Low 32 bits of a generic pointer to __shared__ == LDS byte offset (ISA 10.2).
__device__ __forceinline__ unsigned lds_off(const void* p) {
  return (unsigned)(uintptr_t)p;
}

// One per-lane 16B async copy global->LDS (ASYNCcnt++). Non-temporal.
__device__ __forceinline__ void async_ld16(unsigned lds, const void* g) {
  asm volatile("global_load_async_to_lds_b128 %0, %1, off th:TH_LOAD_NT"
               :: "v"(lds), "v"(g) : "memory");
}

template <int N>
__device__ __forceinline__ void wait_async() {
  asm volatile("s_wait_asynccnt %0" :: "n"(N) : "memory");
}

// Issue the two async ops (src idx + dst idx) for one 64-edge group.
__device__ __forceinline__ void issue_group(unsigned bufbase, int lane,
                                            const long long* src,
                                            const long long* dst,
                                            long long grp) {
  const char* gs = (const char*)(src + grp * EPW) + lane * 16;
  const char* gd = (const char*)(dst + grp * EPW) + lane * 16;
  async_ld16(bufbase + (unsigned)(lane * 16), gs);
  async_ld16(bufbase + 512u + (unsigned)(lane * 16), gd);
}

// agg[dst[e]] += x[src[e]] over all edges.
__global__ __launch_bounds__(THREADS) void scatter_add_kernel(
    const float* __restrict__ x,
    const long long* __restrict__ src,
    const long long* __restrict__ dst,
    float* __restrict__ agg,
    long long nEdges) {
  __shared__ __align__(16) char smem[WAVE_LDS * WPB];

  const int lane      = threadIdx.x & (WAVESZ - 1);
  const int waveInBlk = threadIdx.x >> 5;
  const long long wid = (long long)blockIdx.x * WPB + waveInBlk;
  const long long nw  = (long long)gridDim.x * WPB;

  char* wbase = smem + waveInBlk * WAVE_LDS;
  const unsigned lbase = lds_off(wbase);

  const long long nFull = nEdges / EPW;   // full 64-edge groups

  // ---- prime: issue groups wid + k*nw, k = 0..PIPE-2 into buffers 0..PIPE-2
  int pend = 0;                            // groups issued, not yet consumed
  for (int k = 0; k < PIPE - 1; ++k) {     // wave-uniform conditions
    const long long gk = wid + (long long)k * nw;
    if (gk < nFull) { issue_group(lbase + (unsigned)(k * BUF_BYTES), lane, src, dst, gk); ++pend; }
  }

  // ---- steady state: prefetch group j+PIPE-1 into the buffer freed last iter
  long long j = 0;
  for (long long g = wid; g < nFull; g += nw, ++j) {
    const long long gpre = g + (long long)(PIPE - 1) * nw;
    if (gpre < nFull) {
      issue_group(lbase + (unsigned)(((j + PIPE - 1) & (PIPE - 1)) * BUF_BYTES),
                  lane, src, dst, gpre);
      ++pend;
    }
    // Oldest group done when ASYNCcnt <= 2*(pend-1) (in-order completion).
    switch (pend) {
      case 4: wait_async<6>(); break;
      case 3: wait_async<4>(); break;
      case 2: wait_async<2>(); break;
      default: wait_async<0>(); break;
    }
    // Consume buffer j % PIPE: each lane owns the 2 edges it DMA'd.
    char* cb = wbase + (j & (PIPE - 1)) * BUF_BYTES;
    const longlong2 sp = ((const longlong2*)cb)[lane];          // ds_load_b128
    const longlong2 dp = ((const longlong2*)(cb + 512))[lane];  // ds_load_b128
    const float v0 = x[sp.x];             // random gather, L2-resident (4 MB)
    const float v1 = x[sp.y];
    unsafeAtomicAdd(&agg[dp.x], v0);      // global_atomic_add_f32 @ L2
    unsafeAtomicAdd(&agg[dp.y], v1);
    --pend;
  }

  // ---- tail (nEdges % 64) with plain loads
  const long long tail0 = nFull * EPW;
  for (long long e = tail0 + (long long)blockIdx.x * blockDim.x + threadIdx.x;
       e < nEdges; e += (long long)gridDim.x * blockDim.x) {
    unsafeAtomicAdd(&agg[dst[e]], x[src[e]]);
  }
}

// x1 = relu(agg * w[0]); re-zero agg for the next layer.
__global__ __launch_bounds__(THREADS) void relu_scale_zero_kernel(
    const float* __restrict__ w, float* __restrict__ agg,
    float* __restrict__ xout, int n) {
  const int i = blockIdx.x * blockDim.x + threadIdx.x;
  if (i < n) {
    const float v = agg[i] * w[0];
    xout[i] = v > 0.0f ? v : 0.0f;
    agg[i] = 0.0f;
  }
}

// out = sigmoid(relu(agg * w[1]) + bias)
__global__ __launch_bounds__(THREADS) void final_kernel(
    const float* __restrict__ w, const float* __restrict__ bias,
    const float* __restrict__ agg, float* __restrict__ out, int n) {
  const int i = blockIdx.x * blockDim.x + threadIdx.x;
  if (i < n) {
    float v = agg[i] * w[1];
    v = v > 0.0f ? v : 0.0f;
    const float t = v + bias[0];
    out[i] = 1.0f / (1.0f + expf(-t));
  }
}

extern "C" void kernel_launch(void* const* d_in, const int* in_sizes, int n_in,
                              void* d_out, int out_size, void* d_ws, size_t ws_size,
                              hipStream_t stream) {
  const float*     x    = (const float*)d_in[0];       // [N,1] f32
  const long long* ei   = (const long long*)d_in[1];   // [2,E] i64 row-major
  const float*     w    = (const float*)d_in[2];       // [2] f32
  const float*     bias = (const float*)d_in[3];       // [1] f32
  float* out = (float*)d_out;                          // [N] f32 (also x1 temp)
  float* agg = (float*)d_ws;                           // [N] f32 accumulator

  const int       n  = in_sizes[0];                    // 1,000,000
  const long long nE = (long long)in_sizes[1] / 2;     // 16,000,000
  const long long* src = ei;
  const long long* dst = ei + nE;

  const int sblocks = 1024;                 // 8192 waves, ~30.5 groups each
  const int mblocks = (n + THREADS - 1) / THREADS;

  hipMemsetAsync(agg, 0, (size_t)n * sizeof(float), stream);
  // Layer 0
  scatter_add_kernel<<<sblocks, THREADS, 0, stream>>>(x, src, dst, agg, nE);
  relu_scale_zero_kernel<<<mblocks, THREADS, 0, stream>>>(w, agg, out, n);
  // Layer 1 (reads x1 from d_out, agg already re-zeroed)
  scatter_add_kernel<<<sblocks, THREADS, 0, stream>>>(out, src, dst, agg, nE);
  final_kernel<<<mblocks, THREADS, 0, stream>>>(w, bias, agg, out, n);
}